// TranslationFromAnglesBlock_78185584656835
// MI455X (gfx1250) — compile-verified
//
#include <hip/hip_runtime.h>
#include <hip/hip_bf16.h>

#define TPB   256
#define RPB   1024              // rows per block (4 rows per thread)

typedef __attribute__((ext_vector_type(4))) int v4i;
typedef __attribute__((address_space(1))) v4i gv4i;
typedef __attribute__((address_space(3))) v4i lv4i;

// Gram-Schmidt: return the normalized second (orthogonalized) column.
__device__ __forceinline__ void gs_col2(float x0, float x1, float y0, float y1,
                                        float& o0, float& o1) {
    const float EPS = 1e-12f;
    float nx  = sqrtf(x0 * x0 + x1 * x1);
    float inx = 1.0f / fmaxf(nx, EPS);
    float xh0 = x0 * inx, xh1 = x1 * inx;
    float dot = xh0 * y0 + xh1 * y1;
    float t0  = y0 - xh0 * dot;
    float t1  = y1 - xh1 * dot;
    float nt  = sqrtf(t0 * t0 + t1 * t1);
    float it  = 1.0f / fmaxf(nt, EPS);
    o0 = t0 * it;
    o1 = t1 * it;
}

__device__ __forceinline__ void row_compute(const float f[9],
                                            float& o0, float& o1, float& o2) {
    float yx0, yx1, yz0, yz1;
    gs_col2(f[0], f[1], f[2], f[3], yx0, yx1);   // Rx column 1 = (Rx01, Rx11)
    gs_col2(f[4], f[5], f[6], f[7], yz0, yz1);   // Rz column 1 = (Rz01, Rz11)
    const float d = f[8];
    o0 = yz0 * yx0 * d;   // R[0][2] * d
    o1 = yz1 * yx0 * d;   // R[1][2] * d
    o2 = yx1 * d;         // R[2][2] * d
}

__global__ __launch_bounds__(TPB) void
tfa_kernel(const float* __restrict__ in, float* __restrict__ out, int B) {
    __shared__ float s[RPB * 9];          // 36 KB: staging for in, reused for out

    const int tid = threadIdx.x;
    const long long blockRow = (long long)blockIdx.x * RPB;
    const float* gin  = in  + blockRow * 9;
    float*       gout = out + blockRow * 3;
    const long long rowsLeftLL = (long long)B - blockRow;
    const int rowsLeft = rowsLeftLL > RPB ? RPB : (int)rowsLeftLL;

    if (rowsLeft == RPB) {
        // ---- Full tile: async DMA memory->LDS, 512 B per wave-instruction ----
#pragma unroll
        for (int i = 0; i < 9; ++i) {
            const int e = i * TPB + tid;                 // b128 element index
            __builtin_amdgcn_global_load_async_to_lds_b128(
                (gv4i*)(gin + 4 * e), (lv4i*)(&s[4 * e]), 0, 0);
        }
        __builtin_amdgcn_s_wait_asynccnt(0);
        __syncthreads();

        // ---- Compute 4 rows per thread (LDS stride 9 -> conflict-free) ----
        float o0[4], o1[4], o2[4];
#pragma unroll
        for (int k = 0; k < 4; ++k) {
            const int r = tid + k * TPB;
            float f[9];
#pragma unroll
            for (int i = 0; i < 9; ++i) f[i] = s[r * 9 + i];
            row_compute(f, o0[k], o1[k], o2[k]);
        }
        __syncthreads();   // all reads done before LDS is reused

        // ---- Restage outputs (stride 3 -> conflict-free) ----
#pragma unroll
        for (int k = 0; k < 4; ++k) {
            const int r = tid + k * TPB;
            s[r * 3 + 0] = o0[k];
            s[r * 3 + 1] = o1[k];
            s[r * 3 + 2] = o2[k];
        }
        __syncthreads();

        // ---- Async DMA LDS->memory, 3 b128 per thread ----
#pragma unroll
        for (int i = 0; i < 3; ++i) {
            const int e = i * TPB + tid;
            __builtin_amdgcn_global_store_async_from_lds_b128(
                (gv4i*)(gout + 4 * e), (lv4i*)(&s[4 * e]), 0, 0);
        }
        __builtin_amdgcn_s_wait_asynccnt(0);
    } else {
        // ---- Tail tile: plain guarded per-row path ----
        for (int r = tid; r < rowsLeft; r += TPB) {
            float f[9];
#pragma unroll
            for (int i = 0; i < 9; ++i) f[i] = gin[r * 9 + i];
            float a, b, c;
            row_compute(f, a, b, c);
            gout[r * 3 + 0] = a;
            gout[r * 3 + 1] = b;
            gout[r * 3 + 2] = c;
        }
    }
}

extern "C" void kernel_launch(void* const* d_in, const int* in_sizes, int n_in,
                              void* d_out, int out_size, void* d_ws, size_t ws_size,
                              hipStream_t stream) {
    (void)n_in; (void)out_size; (void)d_ws; (void)ws_size;
    const float* in  = (const float*)d_in[0];
    float*       out = (float*)d_out;
    const long long B = (long long)in_sizes[0] / 9;   // 4194304 rows of 9 floats
    const int grid = (int)((B + RPB - 1) / RPB);      // 4096 blocks
    tfa_kernel<<<grid, TPB, 0, stream>>>(in, out, (int)B);
}